// Model_70471823392989
// MI455X (gfx1250) — compile-verified
//
#include <hip/hip_runtime.h>
#include <stdint.h>

// ---------------------------------------------------------------------------
// Bulk copy of input_ids -> out via the CDNA5 async global<->LDS data mover.
// 128-bit per lane, 4 buffers deep, ASYNCcnt-tracked. Data never touches
// VGPRs. Default (RT) temporal hints: the ~140MB working set fits in the
// 192MB L2, and the harness replays the same graph, so we want L2 residency.
// ---------------------------------------------------------------------------
#define NBUF 4
#define CP_BLOCK 256

__global__ void __launch_bounds__(CP_BLOCK)
copy_async_kernel(const int* __restrict__ in, int* __restrict__ out,
                  long long nvec, long long num_tokens) {
    __shared__ int4 stage[NBUF][CP_BLOCK];   // 16 KB per workgroup
    const int tid = threadIdx.x;
    const long long gsz = (long long)gridDim.x * CP_BLOCK;
    long long i = (long long)blockIdx.x * CP_BLOCK + tid;

    // Per-lane LDS byte addresses of the staging slots. On AMDGPU the low 32
    // bits of a generic pointer into the shared aperture are exactly the
    // workgroup-relative LDS byte address the async ops consume.
    unsigned lds_addr[NBUF];
#pragma unroll
    for (int j = 0; j < NBUF; ++j)
        lds_addr[j] = (unsigned)(uintptr_t)&stage[j][tid];

    const int4* __restrict__ vin = (const int4*)in;
    int4* __restrict__ vout = (int4*)out;

    for (; i < nvec; i += gsz * NBUF) {
#pragma unroll
        for (int j = 0; j < NBUF; ++j) {
            long long idx = i + (long long)j * gsz;
            if (idx < nvec) {
                const int4* g = vin + idx;
                asm volatile("global_load_async_to_lds_b128 %0, %1, off"
                             :: "v"(lds_addr[j]), "v"(g) : "memory");
            }
        }
        asm volatile("s_wait_asynccnt 0" ::: "memory");   // loads landed in LDS
#pragma unroll
        for (int j = 0; j < NBUF; ++j) {
            long long idx = i + (long long)j * gsz;
            if (idx < nvec) {
                int4* g = vout + idx;
                asm volatile("global_store_async_from_lds_b128 %0, %1, off"
                             :: "v"(g), "v"(lds_addr[j]) : "memory");
            }
        }
        asm volatile("s_wait_asynccnt 0" ::: "memory");   // buffers reusable
    }

    // Scalar tail: num_tokens % 4 elements (< 4 total).
    for (long long t = nvec * 4 + (long long)blockIdx.x * CP_BLOCK + tid;
         t < num_tokens; t += gsz)
        out[t] = in[t];
}

// ---------------------------------------------------------------------------
// Per-request kernel: removes the searchsorted entirely.
//  - logits_indices[j] = qsl[r+1] - cu[r+1] + j   for j in [cu[r], cu[r+1])
//  - scatter last_sampled + draft tokens over the tail of the query span
//    when the request is active (seq_lens[r] > prefill_len[idx_mapping[r]]).
// Positions are disjoint across requests, so no atomics needed.
// ---------------------------------------------------------------------------
__global__ void __launch_bounds__(256)
scatter_logits_kernel(const int* __restrict__ idx_mapping,
                      const int* __restrict__ last_sampled,
                      const int* __restrict__ qsl,
                      const int* __restrict__ seq_lens,
                      const int* __restrict__ prefill_len,
                      const int* __restrict__ draft,
                      const int* __restrict__ cu,
                      int* __restrict__ ids_out,
                      int* __restrict__ logits_out,
                      int num_reqs, int spec_steps) {
    int r = blockIdx.x * blockDim.x + threadIdx.x;
    if (r >= num_reqs) return;

    int cu0 = cu[r], cu1 = cu[r + 1];
    int q1 = qsl[r + 1];

    // logits_indices for this request's range
    int base = q1 - cu1;
    for (int j = cu0; j < cu1; ++j)
        logits_out[j] = base + j;

    // token scatter
    int rs = idx_mapping[r];
    if (seq_lens[r] > prefill_len[rs]) {
        int nlog = cu1 - cu0;          // 1..spec_steps+1
        int start = q1 - nlog;
        ids_out[start] = last_sampled[rs];
        for (int k = 1; k < nlog; ++k)
            ids_out[start + k] = draft[rs * spec_steps + (k - 1)];
    }
}

extern "C" void kernel_launch(void* const* d_in, const int* in_sizes, int n_in,
                              void* d_out, int out_size, void* d_ws, size_t ws_size,
                              hipStream_t stream) {
    // setup_inputs() order:
    // 0: input_ids (num_tokens)      1: idx_mapping (NR)
    // 2: last_sampled_tokens (NR)    3: query_start_loc (NR+1)
    // 4: seq_lens (NR)               5: prefill_len (NR)
    // 6: draft_tokens (NR*SPEC)      7: cu_num_logits (NR+1)
    // 8: num_logits (scalar)
    const int* input_ids    = (const int*)d_in[0];
    const int* idx_mapping  = (const int*)d_in[1];
    const int* last_sampled = (const int*)d_in[2];
    const int* qsl          = (const int*)d_in[3];
    const int* seq_lens     = (const int*)d_in[4];
    const int* prefill_len  = (const int*)d_in[5];
    const int* draft        = (const int*)d_in[6];
    const int* cu           = (const int*)d_in[7];

    const long long num_tokens = in_sizes[0];
    const int num_reqs   = in_sizes[1];
    const int spec_steps = in_sizes[6] / num_reqs;

    int* out_ids    = (int*)d_out;
    int* out_logits = out_ids + num_tokens;

    // Copy kernel: one 4-deep b128 burst per thread.
    long long nvec = num_tokens / 4;
    long long per_block = (long long)CP_BLOCK * NBUF;
    int cp_blocks = (int)((nvec + per_block - 1) / per_block);
    if (cp_blocks < 1) cp_blocks = 1;
    copy_async_kernel<<<cp_blocks, CP_BLOCK, 0, stream>>>(
        input_ids, out_ids, nvec, num_tokens);

    // Scatter + logits indices after the copy (stream-ordered).
    int rb = (num_reqs + 255) / 256;
    scatter_logits_kernel<<<rb, 256, 0, stream>>>(
        idx_mapping, last_sampled, qsl, seq_lens, prefill_len, draft, cu,
        out_ids, out_logits, num_reqs, spec_steps);
}